// VoxelConvolution_22093311771084
// MI455X (gfx1250) — compile-verified
//
#include <hip/hip_runtime.h>
#include <hip/hip_bf16.h>

typedef __attribute__((ext_vector_type(2))) float v2f;
typedef __attribute__((ext_vector_type(8))) float v8f;

// N_VOX: lattice points in [-2,2]^3 with |p| <= 2.5  -> 125 - 44 = 81
#define NVOX 81.0f

#define WMMA_F32(A, B, C) \
  __builtin_amdgcn_wmma_f32_16x16x4_f32(false, (A), false, (B), (short)0, (C), false, false)

// ---------------------------------------------------------------------------
// Phase 1: build the 128x128x125 conv kernel from the small weights.
// Output layout is pre-swizzled for the WMMA A-fragment:
//   Kswz[((t*64 + (c>>1))*128 + o)*2 + (c&1)] = kernel[o][c][t]
// so a lane's A operand (K-pair of input channels for its output row) is one
// 8-byte contiguous load in exactly the ISA's 16x4 fp32 A layout.
// ---------------------------------------------------------------------------
__global__ __launch_bounds__(256) void vox_build_kernel(
    const float* __restrict__ weight,   // (8, 4096)  [rb][j*1024 + u*32 + w]
    const float* __restrict__ scs,      // (32, 32)   [u][w]
    const float* __restrict__ scv,      // (32, 32)   [u][w]
    float* __restrict__ Kswz)           // 125*64*128*2 floats (8 MB)
{
  int idx = blockIdx.x * 256 + threadIdx.x;
  if (idx >= 125 * 128 * 128) return;
  int o = idx & 127;
  int c = (idx >> 7) & 127;
  int t = idx >> 14;
  int tz = t / 25, ty = (t / 5) % 5, tx = t % 5;

  float fx = (float)(tz - 2), fy = (float)(ty - 2), fz = (float)(tx - 2);
  float dd = sqrtf(fx * fx + fy * fy + fz * fz);
  float n0 = 0.f, n1 = 0.f, n2 = 0.f;
  if (dd > 1e-12f) { n0 = fx / dd; n1 = fy / dd; n2 = fz / dd; }

  const float step = 2.5f / 9.0f;                 // R / (NUM_RB+1)
  const float SM   = 1.14136f * expf(2.0f);       // SMOOTH_C
  const float PW0  = 0.125f;                      // sqrt(1/64)
  const float PW1  = 0.21650635094610965f;        // sqrt(3/64)
  const float SQ3  = 1.7320508075688772f;

  int j, u, w; float fac;
  if (o < 32 && c < 32)      { j = 0; u = c; w = o; fac = PW0; }                       // Kss
  else if (o < 32)           { int cm = c - 32; j = 3; u = cm / 3; w = o;
                               int i = cm % 3;
                               fac = PW0 * (i == 0 ? n0 : (i == 1 ? n1 : n2)); }       // Ksv
  else if (c < 32)           { int om = o - 32; j = 1; u = c; w = om / 3;
                               int k = om % 3;
                               fac = PW1 * (k == 0 ? n0 : (k == 1 ? n1 : n2)); }       // Kvs
  else                       { int om = o - 32, cm = c - 32; j = 2;
                               u = cm / 3; w = om / 3;
                               fac = ((om % 3) == (cm % 3)) ? (PW1 / SQ3) : 0.0f; }    // Kvv

  float accw = 0.0f;
  #pragma unroll
  for (int rb = 0; rb < 8; ++rb) {
    float v    = (float)(rb + 1) * step;
    float diff = (dd - v) / step;
    float f = 0.0f;
    if (fabsf(diff) < 1.0f) {
      float dc = fminf(fmaxf(diff, -1.0f + 1e-6f), 1.0f - 1e-6f);
      f = SM * expf(-1.0f / (1.0f + dc)) * expf(-1.0f / (1.0f - dc));
    }
    accw += f * weight[rb * 4096 + j * 1024 + u * 32 + w];
  }
  float val = fac * accw / NVOX;

  if (t == 62) {  // center tap (2,2,2): add self-connection / N_VOX
    const float inv = 0.17677669529663687f;  // 1/sqrt(32)
    if (o < 32 && c < 32) {
      val += inv * scs[c * 32 + o] / NVOX;               // sc_s = inv * scs^T
    } else if (o >= 32 && c >= 32) {
      int om = o - 32, cm = c - 32;
      if ((om % 3) == (cm % 3))
        val += inv * scv[(cm / 3) * 32 + (om / 3)] / NVOX;
    }
  }

  Kswz[((t * 64 + (c >> 1)) * 128 + o) * 2 + (c & 1)] = val;
}

// ---------------------------------------------------------------------------
// Phase 2: implicit GEMM conv with V_WMMA_F32_16X16X4_F32.
//   Block: 256 threads = 8 waves; wave i owns output-channel tile [16i,16i+16).
//   Block covers 2 output rows x 48 w-positions  ->  6 position tiles / wave.
//   One A fragment (global_load_b64 from the L2-resident swizzled kernel) is
//   reused by 6 WMMAs, cutting A-side L2 traffic 6x vs. a 1-tile design.
//   B activations staged in LDS (2 rows x 128 ch x 52 w = 52 KB), shared by
//   all 8 waves. K-accumulation: 125 taps x 32 chunks of 4 = 4000 A-steps,
//   24000 WMMAs per wave.
// ---------------------------------------------------------------------------
#define STG_ROW 52            // stage width: 48 + 2*pad(2)
#define STG_PLANE (128 * STG_ROW)

__global__ __launch_bounds__(256) void vox_conv_wmma(
    const float* __restrict__ x,       // (128, 48, 48, 48)
    const float* __restrict__ Kswz,    // swizzled kernel, 8 MB (L2-resident)
    float* __restrict__ out)           // (128, 48, 48, 48)
{
  __shared__ float stage[2 * STG_PLANE];   // [row r][channel c][w 0..51]

  const int lane = threadIdx.x & 31;
  const int wave = threadIdx.x >> 5;
  const int ll   = lane & 15;          // N / M index within half-wave
  const int lh   = lane >> 4;          // half-wave select

  const int bx  = blockIdx.x;          // grid: 24 * 48
  const int oh0 = (bx % 24) * 2;       // two output rows oh0, oh0+1
  const int od  = bx / 24;

  // Per-lane A offset inside one tap's 64x128x2 block:
  //   c-pair index = cc*2 + lh, output row = wave*16 + ll
  const int laneA = (lh * 128 + wave * 16 + ll) * 2;

  v8f acc[6];
  #pragma unroll
  for (int p = 0; p < 6; ++p) acc[p] = (v8f){};

  for (int tz = 0; tz < 5; ++tz) {
    const int id  = od + tz - 2;
    const bool dOK = ((unsigned)id < 48u);
    for (int ty = 0; ty < 5; ++ty) {

      __syncthreads();
      // Stage x[all c][w = -2 .. 49] for the two input rows, zero-padded.
      for (int k2 = threadIdx.x; k2 < 2 * STG_PLANE; k2 += 256) {
        int jw = k2 % STG_ROW;
        int rc = k2 / STG_ROW;
        int cch = rc & 127;
        int r   = rc >> 7;
        int ih  = oh0 + r + ty - 2;
        int iw  = jw - 2;
        float v = 0.0f;
        if (dOK && ((unsigned)ih < 48u) && ((unsigned)iw < 48u))
          v = x[((cch * 48 + id) * 48 + ih) * 48 + iw];
        stage[k2] = v;
      }
      __syncthreads();

      const int tb = tz * 25 + ty * 5;
      for (int tx = 0; tx < 5; ++tx) {
        const float* Abase = Kswz + (size_t)(tb + tx) * (64 * 128 * 2) + laneA;
        // Pull next tap's A block toward the caches (global_prefetch_b8).
        __builtin_prefetch(Abase + 64 * 128 * 2, 0, 0);

        const float* sbase = &stage[(size_t)lh * STG_ROW + tx + ll];

        #pragma unroll 4
        for (int cc = 0; cc < 32; ++cc) {
          // A: kernel[co = wave*16+ll][c = cc*4 + 2*lh + {0,1}][tap]
          v2f a = *(const v2f*)(Abase + cc * 512);
          const float* s0 = sbase + cc * (4 * STG_ROW);
          // 6 position tiles share this A fragment.
          #pragma unroll
          for (int p = 0; p < 6; ++p) {
            const int off = (p >= 3 ? STG_PLANE : 0) + (p % 3) * 16;
            v2f b;
            b.x = s0[off];                 // B row K = lh
            b.y = s0[off + 2 * STG_ROW];   // B row K = lh + 2
            acc[p] = WMMA_F32(a, b, acc[p]);
          }
        }
      }
    }
  }

  // Write the six 16x16 tiles: VGPR q holds Out[wave*16 + 8*lh + q][pos].
  #pragma unroll
  for (int p = 0; p < 6; ++p) {
    const int r  = (p >= 3) ? 1 : 0;
    const int sp = od * 2304 + (oh0 + r) * 48 + (p % 3) * 16 + ll;
    #pragma unroll
    for (int q = 0; q < 8; ++q) {
      int o = wave * 16 + 8 * lh + q;
      out[o * 110592 + sp] = acc[p][q];
    }
  }
}

// ---------------------------------------------------------------------------
extern "C" void kernel_launch(void* const* d_in, const int* in_sizes, int n_in,
                              void* d_out, int out_size, void* d_ws, size_t ws_size,
                              hipStream_t stream) {
  const float* x      = (const float*)d_in[0];  // (1,128,48,48,48)
  const float* weight = (const float*)d_in[1];  // (8, 4096)
  const float* scs    = (const float*)d_in[2];  // (32, 32)
  const float* scv    = (const float*)d_in[3];  // (32, 32)
  float* out  = (float*)d_out;
  float* Kswz = (float*)d_ws;                   // needs 8 MB scratch

  const int total = 125 * 128 * 128;
  vox_build_kernel<<<(total + 255) / 256, 256, 0, stream>>>(weight, scs, scv, Kswz);

  // 24*48 = 1152 blocks; each computes 128 co x (2 rows x 48) positions.
  vox_conv_wmma<<<24 * 48, 256, 0, stream>>>(x, Kswz, out);
}